// BarycentricPooling_60928406061261
// MI455X (gfx1250) — compile-verified
//
#include <hip/hip_runtime.h>
#include <hip/hip_bf16.h>
#include <math.h>

#define HIDDEN  256
#define CB_N    16
#define DIST    8
#define NGRAPH  128
#define EPS_INV 10.0f     // 1/0.1
#define ITERS   20
#define CBS     260       // LDS codebook row stride (padded, conflict-free)

typedef float v2f __attribute__((ext_vector_type(2)));
typedef float v8f __attribute__((ext_vector_type(8)));

// ---------------------------------------------------------------------------
// Init: zero counts & all 3 S buffers, ynorm[j] = sum_k codebook[j][k]^2
// ---------------------------------------------------------------------------
__global__ void k_init(const float* __restrict__ cb, float* __restrict__ Sbuf,
                       float* __restrict__ counts, float* __restrict__ ynorm) {
    int t = threadIdx.x;
    for (int i = t; i < 3 * NGRAPH * CB_N; i += blockDim.x) Sbuf[i] = 0.0f;
    for (int i = t; i < NGRAPH; i += blockDim.x) counts[i] = 0.0f;
    if (t < CB_N) {
        float s = 0.0f;
        for (int k = 0; k < HIDDEN; ++k) { float c = cb[t * HIDDEN + k]; s = fmaf(c, c, s); }
        ynorm[t] = s;
    }
}

// ---------------------------------------------------------------------------
// Per-graph counts of flattened points (each node contributes DIST points)
// ---------------------------------------------------------------------------
__global__ void k_counts(const int* __restrict__ bidx, float* __restrict__ counts, int N) {
    __shared__ float h[NGRAPH];
    for (int i = threadIdx.x; i < NGRAPH; i += blockDim.x) h[i] = 0.0f;
    __syncthreads();
    for (int i = blockIdx.x * blockDim.x + threadIdx.x; i < N; i += gridDim.x * blockDim.x)
        atomicAdd(&h[bidx[i]], (float)DIST);
    __syncthreads();
    for (int i = threadIdx.x; i < NGRAPH; i += blockDim.x)
        if (h[i] != 0.0f) atomicAdd(&counts[i], h[i]);
}

// ---------------------------------------------------------------------------
// K = exp(-max(xn + yn - 2*x.cb^T, 0)/eps)  via V_WMMA_F32_16X16X4_F32.
// One wave per 16-point tile; 8 waves (256 threads) per block.
// A: lane l<16 -> x[row l][k..k+1], lane l+16 -> x[row l][k+2..k+3]
// B: same pattern from codebook rows (B[k][n] = cb[n][k]) staged in LDS.
// ---------------------------------------------------------------------------
__global__ void k_computeK(const float* __restrict__ x, const float* __restrict__ cb,
                           const float* __restrict__ ynorm, float* __restrict__ Kout,
                           int Nf, int tiles) {
    __shared__ float lcb[CB_N * CBS];
    for (int i = threadIdx.x; i < CB_N * HIDDEN; i += blockDim.x) {
        int r = i >> 8, c = i & (HIDDEN - 1);
        lcb[r * CBS + c] = cb[i];
    }
    __syncthreads();

    int wid  = threadIdx.x >> 5;
    int tile = blockIdx.x * (blockDim.x >> 5) + wid;
    if (tile >= tiles) return;

    int lane = threadIdx.x & 31;
    int half = lane >> 4;          // 0: K+{0,1}, 1: K+{2,3}
    int m    = lane & 15;          // A row / B column index for this lane

    int rowbase = tile * 16;
    int rA = rowbase + m; if (rA > Nf - 1) rA = Nf - 1;
    const float* __restrict__ xrow = x + (size_t)rA * HIDDEN + 2 * half;
    const float* __restrict__ brow = &lcb[m * CBS + 2 * half];

    v8f acc = {};
    float xnp = 0.0f;
    #pragma unroll 8
    for (int k = 0; k < HIDDEN; k += 4) {
        float2 af = *(const float2*)(xrow + k);
        float2 bf = *(const float2*)(brow + k);
        v2f a; a.x = af.x; a.y = af.y;
        v2f b; b.x = bf.x; b.y = bf.y;
        xnp = fmaf(af.x, af.x, xnp);
        xnp = fmaf(af.y, af.y, xnp);
        acc = __builtin_amdgcn_wmma_f32_16x16x4_f32(
                  false, a, false, b, (short)0, acc, false, false);
    }
    float xnorm_row = xnp + __shfl_xor(xnp, 16, 32);  // lanes l, l^16 share a row
    float yn = ynorm[m];                              // column norm, n = lane&15

    #pragma unroll
    for (int r = 0; r < 8; ++r) {
        int mo = r + 8 * half;                        // output row within tile
        float xn = __shfl(xnorm_row, mo, 32);         // row mo's norm lives at lane mo
        float cost = xn + yn - 2.0f * acc[r];
        cost = fmaxf(cost, 0.0f);
        float Kv = __expf(-cost * EPS_INV);
        int row = rowbase + mo;
        if (row < Nf) Kout[(size_t)row * CB_N + m] = Kv;
    }
}

// ---------------------------------------------------------------------------
// Fused Sinkhorn iteration (triple-buffered S):
//   v      = (1/CB)/(Sprev + 1e-8)            (it==0: v = 1)   -> LDS table
//   u_i    = (1/counts[seg])/(K_i . v[seg] + 1e-8)
//   Snext[b,:] += K_i * u_i                   (LDS block-reduce, then atomics)
//   block 0 zeroes Szero (the buffer iteration it+1 will accumulate into)
// K (25.6 MB) stays resident in the 192 MB L2 across all 20 iterations.
// ---------------------------------------------------------------------------
__global__ void k_iter(const float* __restrict__ Kmat, const int* __restrict__ bidx,
                       const float* __restrict__ counts,
                       const float* __restrict__ Sprev, float* __restrict__ Snext,
                       float* __restrict__ Szero, int it, int Nf) {
    __shared__ float acc[NGRAPH * CB_N];
    __shared__ float vtab[NGRAPH * CB_N];
    __shared__ float rc[NGRAPH];

    for (int i = threadIdx.x; i < NGRAPH * CB_N; i += blockDim.x) {
        acc[i]  = 0.0f;
        vtab[i] = (it == 0) ? 1.0f : (1.0f / CB_N) / (Sprev[i] + 1e-8f);
    }
    for (int i = threadIdx.x; i < NGRAPH; i += blockDim.x) rc[i] = 1.0f / counts[i];
    if (blockIdx.x == 0)
        for (int i = threadIdx.x; i < NGRAPH * CB_N; i += blockDim.x) Szero[i] = 0.0f;
    __syncthreads();

    for (int i = blockIdx.x * blockDim.x + threadIdx.x; i < Nf; i += gridDim.x * blockDim.x) {
        int seg = bidx[i >> 3];   // DIST == 8
        const float4* Kr = (const float4*)(Kmat + (size_t)i * CB_N);
        const float4* vr = (const float4*)(&vtab[seg * CB_N]);
        float4 k0 = Kr[0], k1 = Kr[1], k2 = Kr[2], k3 = Kr[3];
        float4 v0 = vr[0], v1 = vr[1], v2 = vr[2], v3 = vr[3];
        float s = k0.x * v0.x + k0.y * v0.y + k0.z * v0.z + k0.w * v0.w
                + k1.x * v1.x + k1.y * v1.y + k1.z * v1.z + k1.w * v1.w
                + k2.x * v2.x + k2.y * v2.y + k2.z * v2.z + k2.w * v2.w
                + k3.x * v3.x + k3.y * v3.y + k3.z * v3.z + k3.w * v3.w;
        float u = rc[seg] / (s + 1e-8f);
        float* ab = &acc[seg * CB_N];
        atomicAdd(&ab[0],  k0.x * u); atomicAdd(&ab[1],  k0.y * u);
        atomicAdd(&ab[2],  k0.z * u); atomicAdd(&ab[3],  k0.w * u);
        atomicAdd(&ab[4],  k1.x * u); atomicAdd(&ab[5],  k1.y * u);
        atomicAdd(&ab[6],  k1.z * u); atomicAdd(&ab[7],  k1.w * u);
        atomicAdd(&ab[8],  k2.x * u); atomicAdd(&ab[9],  k2.y * u);
        atomicAdd(&ab[10], k2.z * u); atomicAdd(&ab[11], k2.w * u);
        atomicAdd(&ab[12], k3.x * u); atomicAdd(&ab[13], k3.y * u);
        atomicAdd(&ab[14], k3.z * u); atomicAdd(&ab[15], k3.w * u);
    }
    __syncthreads();
    for (int i = threadIdx.x; i < NGRAPH * CB_N; i += blockDim.x) {
        float val = acc[i];
        if (val != 0.0f) atomicAdd(&Snext[i], val);
    }
}

// ---------------------------------------------------------------------------
// v_final = (1/CB)/(S_final + 1e-8);  weights = normalize(S_final * v_final)
// ---------------------------------------------------------------------------
__global__ void k_final(const float* __restrict__ S, float* __restrict__ out) {
    int b = blockIdx.x * blockDim.x + threadIdx.x;
    if (b < NGRAPH) {
        float t[CB_N]; float d = 0.0f;
        #pragma unroll
        for (int j = 0; j < CB_N; ++j) {
            float s  = S[b * CB_N + j];
            float vv = (1.0f / CB_N) / (s + 1e-8f);
            t[j] = s * vv;
            d += t[j];
        }
        float inv = (d > 0.0f) ? (1.0f / d) : 0.0f;
        #pragma unroll
        for (int j = 0; j < CB_N; ++j) out[b * CB_N + j] = (d > 0.0f) ? t[j] * inv : 0.0f;
    }
}

extern "C" void kernel_launch(void* const* d_in, const int* in_sizes, int n_in,
                              void* d_out, int out_size, void* d_ws, size_t ws_size,
                              hipStream_t stream) {
    (void)n_in; (void)out_size; (void)ws_size;
    const float* x    = (const float*)d_in[0];   // [N, DIST, HIDDEN] f32
    const int*   bidx = (const int*)d_in[1];     // [N] int32
    const float* cb   = (const float*)d_in[2];   // [CB, HIDDEN] f32
    float* out = (float*)d_out;                  // [NGRAPH, CB]

    int N  = in_sizes[1];
    int Nf = N * DIST;

    // workspace layout (floats)
    float* ws     = (float*)d_ws;
    float* Kmat   = ws;                                  // Nf*CB
    float* counts = Kmat + (size_t)Nf * CB_N;            // NGRAPH
    float* Sbuf   = counts + NGRAPH;                     // 3 * NGRAPH*CB (triple buffer)
    float* ynorm  = Sbuf + 3 * NGRAPH * CB_N;            // CB

    const int SB = NGRAPH * CB_N;

    k_init<<<1, 256, 0, stream>>>(cb, Sbuf, counts, ynorm);
    k_counts<<<200, 256, 0, stream>>>(bidx, counts, N);

    int tiles   = (Nf + 15) / 16;
    int blocksK = (tiles + 7) / 8;                       // 8 waves (tiles) per 256-thread block
    k_computeK<<<blocksK, 256, 0, stream>>>(x, cb, ynorm, Kmat, Nf, tiles);

    for (int it = 0; it < ITERS; ++it) {
        k_iter<<<256, 512, 0, stream>>>(Kmat, bidx, counts,
                                        Sbuf + (it % 3) * SB,
                                        Sbuf + ((it + 1) % 3) * SB,
                                        Sbuf + ((it + 2) % 3) * SB,
                                        it, Nf);
    }
    k_final<<<1, 128, 0, stream>>>(Sbuf + (ITERS % 3) * SB, out);
}